// FlowWarpMask_87316685128075
// MI455X (gfx1250) — compile-verified
//
#include <hip/hip_runtime.h>

// Forward-warp coverage mask, tuned for MI455X (gfx1250):
//  - per-tile LDS accumulation (ds_add_f32) with 24px halo (flow sigma = 8px)
//  - CDNA5 async global->LDS copy for flow staging, overlapped with LDS zeroing
//  - clustered global_atomic_add_f32 flush (32 MiB acc lives in 192 MB L2)
//  - float4 zero + threshold passes (pure streaming)

#define BATCH   8
#define IMG_H   1024
#define IMG_W   1024
#define TW      128
#define TH      64
#define HALO    24
#define WIN_W   (TW + 2*HALO)     // 176
#define WIN_H   (TH + 2*HALO)     // 112
#define WIN     (WIN_W * WIN_H)   // 19712 floats
#define TILE_PX (TW * TH)         // 8192 pixels
#define THREADS 256

__global__ void __launch_bounds__(THREADS)
splat_kernel(const float* __restrict__ flow, float* __restrict__ acc)
{
    extern __shared__ float smem[];
    float* s_acc  = smem;           // WIN floats (accumulator window)
    float* s_flow = smem + WIN;     // 2*TILE_PX floats (flow x-plane, y-plane)

    const int tid = threadIdx.x;
    const int x0  = blockIdx.x * TW;
    const int y0  = blockIdx.y * TH;
    const int b   = blockIdx.z;
    const int wx0 = x0 - HALO;
    const int wy0 = y0 - HALO;

    // ---- Kick off async copy of this tile's flow (64 KB) into LDS ----
    // b128 per lane per issue: (2*8192/4)/256 = 16 issues per thread.
    {
        const size_t plane_stride = (size_t)IMG_H * IMG_W;
        #pragma unroll
        for (int k = 0; k < (2 * TILE_PX / 4) / THREADS; ++k) {
            int f4  = (tid + k * THREADS) * 4;        // float idx in concat planes
            int pl  = (f4 >= TILE_PX) ? 1 : 0;
            int t   = f4 - pl * TILE_PX;
            int row = t >> 7;                         // TW == 128
            int col = t & (TW - 1);
            const float* g = flow + (size_t)(b * 2 + pl) * plane_stride
                                  + (size_t)(y0 + row) * IMG_W + (x0 + col);
            unsigned lds = (unsigned)(uintptr_t)(&s_flow[f4]);  // low 32b = LDS offset
            asm volatile("global_load_async_to_lds_b128 %0, %1, off"
                         :: "v"(lds), "v"(g) : "memory");
        }
    }

    // ---- Zero the LDS window while the async copy is in flight ----
    for (int j = tid; j < WIN; j += THREADS) s_acc[j] = 0.0f;

    asm volatile("s_wait_asynccnt 0x0" ::: "memory");
    __syncthreads();

    // ---- Splat: 32 pixels per thread ----
    #pragma unroll 4
    for (int i = 0; i < TILE_PX / THREADS; ++i) {
        int p  = tid + i * THREADS;
        int px = p & (TW - 1);
        int py = p >> 7;
        float fl_x = s_flow[p];
        float fl_y = s_flow[TILE_PX + p];
        float tx = (float)(x0 + px) + fl_x;
        float ty = (float)(y0 + py) + fl_y;
        float fx0 = floorf(tx), fy0 = floorf(ty);
        float fx = tx - fx0,    fy = ty - fy0;
        int   xi = (int)fx0,    yi = (int)fy0;
        float w00 = (1.0f - fx) * (1.0f - fy);
        float w10 = fx * (1.0f - fy);
        float w01 = (1.0f - fx) * fy;
        float w11 = fx * fy;

        #define CORNER(XI, YI, WV) do {                                          \
            int _x = (XI), _y = (YI); float _w = (WV);                           \
            if ((unsigned)_x < IMG_W && (unsigned)_y < IMG_H && _w != 0.0f) {    \
                int _wx = _x - wx0, _wy = _y - wy0;                              \
                if ((unsigned)_wx < WIN_W && (unsigned)_wy < WIN_H)              \
                    atomicAdd(&s_acc[_wy * WIN_W + _wx], _w);                    \
                else                                                             \
                    atomicAdd(&acc[((size_t)b * IMG_H + _y) * IMG_W + _x], _w);  \
            } } while (0)

        CORNER(xi,     yi,     w00);
        CORNER(xi + 1, yi,     w10);
        CORNER(xi,     yi + 1, w01);
        CORNER(xi + 1, yi + 1, w11);
        #undef CORNER
    }

    __syncthreads();

    // ---- Flush nonzero window cells to the global accumulator ----
    for (int j = tid; j < WIN; j += THREADS) {
        float v = s_acc[j];
        if (v != 0.0f) {
            int wy = j / WIN_W;
            int wx = j - wy * WIN_W;
            int gx = wx0 + wx;
            int gy = wy0 + wy;
            if ((unsigned)gx < IMG_W && (unsigned)gy < IMG_H)
                atomicAdd(&acc[((size_t)b * IMG_H + gy) * IMG_W + gx], v);
        }
    }
}

__global__ void zero_kernel(float4* __restrict__ out, int n4)
{
    int i = blockIdx.x * blockDim.x + threadIdx.x;
    if (i < n4) out[i] = make_float4(0.0f, 0.0f, 0.0f, 0.0f);
}

__global__ void thresh_kernel(float4* __restrict__ out, int n4)
{
    int i = blockIdx.x * blockDim.x + threadIdx.x;
    if (i < n4) {
        float4 v = out[i];
        v.x = (v.x > 1e-5f) ? 1.0f : 0.0f;
        v.y = (v.y > 1e-5f) ? 1.0f : 0.0f;
        v.z = (v.z > 1e-5f) ? 1.0f : 0.0f;
        v.w = (v.w > 1e-5f) ? 1.0f : 0.0f;
        out[i] = v;
    }
}

extern "C" void kernel_launch(void* const* d_in, const int* in_sizes, int n_in,
                              void* d_out, int out_size, void* d_ws, size_t ws_size,
                              hipStream_t stream)
{
    const float* flow = (const float*)d_in[0];   // [B,2,H,W] float32
    float* out = (float*)d_out;                  // [B,1,H,W] -> bool as float 0/1

    // out_size = 8*1024*1024, multiple of 4
    int n4 = out_size >> 2;
    dim3 bt(THREADS), gt((n4 + THREADS - 1) / THREADS);

    zero_kernel<<<gt, bt, 0, stream>>>((float4*)out, n4);

    dim3 grid(IMG_W / TW, IMG_H / TH, BATCH);    // (8,16,8) = 1024 workgroups
    size_t smem_bytes = (size_t)(WIN + 2 * TILE_PX) * sizeof(float);  // 144 KB
    splat_kernel<<<grid, THREADS, smem_bytes, stream>>>(flow, out);

    thresh_kernel<<<gt, bt, 0, stream>>>((float4*)out, n4);
}